// EAEF_65910568125047
// MI455X (gfx1250) — compile-verified
//
#include <hip/hip_runtime.h>
#include <math.h>

#define BATCH 2
#define CCH   256
#define HH    48
#define WW    48
#define HW    (HH*WW)        // 2304
#define K9    9
#define CK    (CCH*K9)       // 2304 (GEMM K)
#define NTOT  HW             // GEMM N per batch
#define KDIM  CK
#define NEL   (BATCH*CCH*HW) // 1,179,648

typedef float v2f __attribute__((ext_vector_type(2)));
typedef float v8f __attribute__((ext_vector_type(8)));

__device__ __forceinline__ float sigmoidf_(float x) { return 1.0f / (1.0f + expf(-x)); }
__device__ __forceinline__ float geluf_(float x)    { return 0.5f * x * (1.0f + erff(x * 0.7071067811865475f)); }

// ---------------- offset conv: 3x3, C=256 -> 27, pad 1 ----------------
__global__ void offset_conv_kernel(const float* __restrict__ x, const float* __restrict__ ow,
                                   const float* __restrict__ ob, float* __restrict__ om) {
    int idx = blockIdx.x * blockDim.x + threadIdx.x;
    if (idx >= BATCH * 27 * HW) return;
    int hw = idx % HW;
    int oc = (idx / HW) % 27;
    int b  = idx / (27 * HW);
    int h = hw / WW, w = hw % WW;
    float acc = ob[oc];
    const float* xb   = x  + (size_t)b * CCH * HW;
    const float* wrow = ow + (size_t)oc * CCH * 9;
    for (int c = 0; c < CCH; ++c) {
        const float* xc = xb + (size_t)c * HW;
        const float* wc = wrow + c * 9;
        #pragma unroll
        for (int t = 0; t < 9; ++t) {
            int yy = h + t / 3 - 1, xx = w + t % 3 - 1;
            if (yy >= 0 && yy < HH && xx >= 0 && xx < WW)
                acc += wc[t] * xc[yy * WW + xx];
        }
    }
    om[idx] = acc;
}

// -------- deformable bilinear sample -> v[b, c*9+k, hw] (im2col) --------
__global__ void build_v_kernel(const float* __restrict__ x, const float* __restrict__ om,
                               float* __restrict__ v) {
    int idx = blockIdx.x * blockDim.x + threadIdx.x;
    if (idx >= BATCH * K9 * CCH * HW) return;
    int hw = idx % HW;
    int c  = (idx / HW) % CCH;
    int k  = (idx / (HW * CCH)) % K9;
    int b  = idx / (HW * CCH * K9);
    int h = hw / WW, w = hw % WW;
    const float* omb = om + (size_t)b * 27 * HW;
    // offset.reshape(B,K,2,H,W): off_y(k)=channel 2k, off_x(k)=channel 2k+1
    float offy = omb[(2 * k)     * HW + hw];
    float offx = omb[(2 * k + 1) * HW + hw];
    float mk   = sigmoidf_(omb[(18 + k) * HW + hw]);
    float py = (float)h + (float)(k / 3 - 1) + offy;
    float px = (float)w + (float)(k % 3 - 1) + offx;
    float y0f = floorf(py), x0f = floorf(px);
    float wy = py - y0f, wx = px - x0f;
    int y0 = (int)y0f, x0 = (int)x0f;
    const float* xc = x + ((size_t)b * CCH + c) * HW;
    float v00 = 0.f, v01 = 0.f, v10 = 0.f, v11 = 0.f;
    bool yi0 = (y0 >= 0) && (y0 < HH),     yi1 = (y0 + 1 >= 0) && (y0 + 1 < HH);
    bool xi0 = (x0 >= 0) && (x0 < WW),     xi1 = (x0 + 1 >= 0) && (x0 + 1 < WW);
    if (yi0 && xi0) v00 = xc[y0 * WW + x0];
    if (yi0 && xi1) v01 = xc[y0 * WW + x0 + 1];
    if (yi1 && xi0) v10 = xc[(y0 + 1) * WW + x0];
    if (yi1 && xi1) v11 = xc[(y0 + 1) * WW + x0 + 1];
    float val = v00 * (1.f - wy) * (1.f - wx) + v01 * (1.f - wy) * wx
              + v10 * wy * (1.f - wx)         + v11 * wy * wx;
    v[((size_t)b * CK + (c * K9 + k)) * HW + hw] = val * mk;
}

// ------------- fp32 WMMA GEMM: Out[b] = Wm(256x2304) x V[b](2304x2304) + bias -------------
// 64x64 block tile, 8 waves; wave -> 16(M) x 32(N); K step 4 via V_WMMA_F32_16X16X4_F32
__global__ __launch_bounds__(256) void wmma_gemm_kernel(
    const float* __restrict__ Wm, const float* __restrict__ V,
    const float* __restrict__ bias, float* __restrict__ Out) {
    int b    = blockIdx.z;
    int tid  = threadIdx.x;
    int wave = tid >> 5, lane = tid & 31;
    int wm = wave & 3, wn = wave >> 2;
    int half = lane >> 4, l15 = lane & 15;
    int kh = half * 2;
    int mrow = blockIdx.y * 64 + wm * 16 + l15;
    int col0 = blockIdx.x * 64 + wn * 32 + l15;
    int col1 = col0 + 16;
    const float* Vb = V + (size_t)b * KDIM * NTOT;
    float*       Ob = Out + (size_t)b * CCH * NTOT;
    v8f acc0 = {}; v8f acc1 = {};
    for (int k0 = 0; k0 < KDIM; k0 += 4) {
        // A 16x4: lane<16 holds (M=l15, K=k0..k0+1), lane>=16 holds K=k0+2..k0+3
        const float* wp = Wm + (size_t)mrow * KDIM + k0 + kh;
        v2f a; a.x = wp[0]; a.y = wp[1];
        // B 4x16: vgpr0 = rows {k0+kh}, vgpr1 = rows {k0+kh+1}, cols across lanes
        const float* vp = Vb + (size_t)(k0 + kh) * NTOT;
        v2f b0, b1;
        b0.x = vp[col0]; b0.y = vp[NTOT + col0];
        b1.x = vp[col1]; b1.y = vp[NTOT + col1];
        acc0 = __builtin_amdgcn_wmma_f32_16x16x4_f32(false, a, false, b0, (short)0, acc0, false, false);
        acc1 = __builtin_amdgcn_wmma_f32_16x16x4_f32(false, a, false, b1, (short)0, acc1, false, false);
    }
    // C/D layout: VGPR r -> (M = base + r + half*8, N = l15)
    int rbase = blockIdx.y * 64 + wm * 16 + half * 8;
    #pragma unroll
    for (int r = 0; r < 8; ++r) {
        int row = rbase + r;
        float bv = bias[row];
        Ob[(size_t)row * NTOT + col0] = acc0[r] + bv;
        Ob[(size_t)row * NTOT + col1] = acc1[r] + bv;
    }
}

// ---------------- BN stats over (B,H,W) per channel ----------------
__global__ void bn_stats_kernel(const float* __restrict__ y,
                                float* __restrict__ mean, float* __restrict__ var) {
    int c = blockIdx.x;
    __shared__ float ss[256], sq[256];
    float s = 0.f, q = 0.f;
    for (int i = threadIdx.x; i < BATCH * HW; i += 256) {
        int b = i / HW, hw = i % HW;
        float x = y[((size_t)b * CCH + c) * HW + hw];
        s += x; q += x * x;
    }
    ss[threadIdx.x] = s; sq[threadIdx.x] = q;
    __syncthreads();
    for (int st = 128; st > 0; st >>= 1) {
        if (threadIdx.x < st) { ss[threadIdx.x] += ss[threadIdx.x + st]; sq[threadIdx.x] += sq[threadIdx.x + st]; }
        __syncthreads();
    }
    if (threadIdx.x == 0) {
        float m = ss[0] / (float)(BATCH * HW);
        mean[c] = m;
        var[c]  = sq[0] / (float)(BATCH * HW) - m * m;
    }
}

__global__ void bn_silu_kernel(const float* __restrict__ yin, float* __restrict__ yout,
                               const float* __restrict__ mean, const float* __restrict__ var,
                               const float* __restrict__ gamma, const float* __restrict__ beta) {
    int idx = blockIdx.x * blockDim.x + threadIdx.x;
    if (idx >= NEL) return;
    int c = (idx / HW) % CCH;
    float x = yin[idx];
    float yv = (x - mean[c]) * rsqrtf(var[c] + 1e-5f) * gamma[c] + beta[c];
    yout[idx] = yv * sigmoidf_(yv);
}

// ---------------- fusion glue kernels ----------------
__global__ void mean_pool_kernel(const float* __restrict__ x, float* __restrict__ out) {
    int idx = blockIdx.x * blockDim.x + threadIdx.x;
    if (idx >= BATCH * CCH) return;
    const float* p = x + (size_t)idx * HW;
    float s = 0.f;
    for (int i = 0; i < HW; ++i) s += p[i];
    out[idx] = s / (float)HW;
}

__global__ void linear_kernel(const float* __restrict__ in, const float* __restrict__ Wm,
                              const float* __restrict__ bias, float* __restrict__ out,
                              int nb, int din, int dout, int act) {
    int idx = blockIdx.x * blockDim.x + threadIdx.x;
    if (idx >= nb * dout) return;
    int j = idx % dout, b = idx / dout;
    const float* ip = in + (size_t)b * din;
    const float* wp = Wm + (size_t)j * din;
    float s = bias[j];
    for (int i = 0; i < din; ++i) s += wp[i] * ip[i];
    if (act == 1)      s = geluf_(s);
    else if (act == 2) s = sigmoidf_(s);
    out[idx] = s;
}

__global__ void norm_cross_kernel(const float* __restrict__ ry, const float* __restrict__ ty,
                                  float* __restrict__ cross) {
    int b = blockIdx.x, t = threadIdx.x;
    __shared__ float sr[256], st[256];
    float r  = ry[b * CCH + t];
    float tv = ty[b * CCH + t];
    sr[t] = r * r; st[t] = tv * tv;
    __syncthreads();
    for (int s = 128; s > 0; s >>= 1) {
        if (t < s) { sr[t] += sr[t + s]; st[t] += st[t + s]; }
        __syncthreads();
    }
    float nr = sqrtf(sr[0]), nt = sqrtf(st[0]);
    cross[b * CCH + t] = sigmoidf_((float)CCH * (r / nr) * (tv / nt));
}

// depthwise-ish 7x7: groups=256, 2 in / 2 out channels per group; input = [Rd*cross, Td*cross]
__global__ void dwconv_kernel(const float* __restrict__ Rd, const float* __restrict__ Td,
                              const float* __restrict__ cross,
                              const float* __restrict__ wdw, const float* __restrict__ bdw,
                              float* __restrict__ dw) {
    int idx = blockIdx.x * blockDim.x + threadIdx.x;
    if (idx >= BATCH * 2 * CCH * HW) return;
    int hw = idx % HW;
    int o  = (idx / HW) % (2 * CCH);
    int b  = idx / (2 * CCH * HW);
    int h = hw / WW, w = hw % WW;
    int g = o >> 1;
    float acc = bdw[o];
    #pragma unroll
    for (int i = 0; i < 2; ++i) {
        int ch = g * 2 + i;
        int cc = ch & (CCH - 1);
        const float* src = (ch < CCH ? Rd : Td) + ((size_t)b * CCH + cc) * HW;
        float sc = cross[b * CCH + cc];
        const float* wp = wdw + ((size_t)o * 2 + i) * 49;
        for (int ky = 0; ky < 7; ++ky) {
            int yy = h + ky - 3;
            if (yy < 0 || yy >= HH) continue;
            for (int kx = 0; kx < 7; ++kx) {
                int xx = w + kx - 3;
                if (xx < 0 || xx >= WW) continue;
                acc += wp[ky * 7 + kx] * src[yy * WW + xx] * sc;
            }
        }
    }
    dw[idx] = acc;
}

__global__ void max_pool_kernel(const float* __restrict__ dw, float* __restrict__ mx) {
    int idx = blockIdx.x * blockDim.x + threadIdx.x;
    if (idx >= BATCH * 2 * CCH) return;
    const float* p = dw + (size_t)idx * HW;
    float m = p[0];
    for (int i = 1; i < HW; ++i) m = fmaxf(m, p[i]);
    mx[idx] = m;
}

// nR = Rd*(cross*fgR + (1-cross)), nT = Td*(cross*fgT + (1-cross))
__global__ void fuse_nrt_kernel(const float* __restrict__ Rd, const float* __restrict__ Td,
                                const float* __restrict__ cross, const float* __restrict__ fg,
                                float* __restrict__ nR, float* __restrict__ nT) {
    int idx = blockIdx.x * blockDim.x + threadIdx.x;
    if (idx >= NEL) return;
    int c = (idx / HW) % CCH;
    int b = idx / (CCH * HW);
    float cr = cross[b * CCH + c];
    float ad = 1.f - cr;
    nR[idx] = Rd[idx] * (cr * fg[b * 2 * CCH + c] + ad);
    nT[idx] = Td[idx] * (cr * fg[b * 2 * CCH + CCH + c] + ad);
}

__global__ void sse_att_kernel(const float* __restrict__ nR, const float* __restrict__ nT,
                               const float* __restrict__ srw, const float* __restrict__ srb,
                               const float* __restrict__ stw, const float* __restrict__ stb,
                               float* __restrict__ att) {
    int idx = blockIdx.x * blockDim.x + threadIdx.x;
    if (idx >= BATCH * HW) return;
    int hw = idx % HW, b = idx / HW;
    float ar = srb[0], at = stb[0];
    const float* rp = nR + (size_t)b * CCH * HW + hw;
    const float* tp = nT + (size_t)b * CCH * HW + hw;
    for (int c = 0; c < CCH; ++c) {
        ar += srw[c] * rp[(size_t)c * HW];
        at += stw[c] * tp[(size_t)c * HW];
    }
    float m  = fmaxf(ar, at);
    float e0 = expf(ar - m), e1 = expf(at - m);
    float inv = 1.f / (e0 + e1);
    att[idx]              = e0 * inv;
    att[BATCH * HW + idx] = e1 * inv;
}

__global__ void apply_att_kernel(float* __restrict__ nR, float* __restrict__ nT,
                                 const float* __restrict__ att) {
    int idx = blockIdx.x * blockDim.x + threadIdx.x;
    if (idx >= NEL) return;
    int hw = idx % HW;
    int b  = idx / (CCH * HW);
    nR[idx] *= att[b * HW + hw];
    nT[idx] *= att[BATCH * HW + b * HW + hw];
}

__global__ void add_kernel(const float* __restrict__ a, const float* __restrict__ b,
                           float* __restrict__ c, int n) {
    int idx = blockIdx.x * blockDim.x + threadIdx.x;
    if (idx < n) c[idx] = a[idx] + b[idx];
}

// ---------------- host-side DCN pipeline ----------------
static void run_dcn(const float* x, const float* w, const float* bias,
                    const float* ow, const float* ob,
                    const float* gamma, const float* beta,
                    float* out, float* ws_v, float* ws_om, float* ws_y,
                    float* ws_mean, float* ws_var, hipStream_t stream) {
    offset_conv_kernel<<<(BATCH * 27 * HW + 255) / 256, 256, 0, stream>>>(x, ow, ob, ws_om);
    build_v_kernel<<<(BATCH * K9 * CCH * HW + 255) / 256, 256, 0, stream>>>(x, ws_om, ws_v);
    dim3 g(NTOT / 64, CCH / 64, BATCH);
    wmma_gemm_kernel<<<g, 256, 0, stream>>>(w, ws_v, bias, ws_y);
    bn_stats_kernel<<<CCH, 256, 0, stream>>>(ws_y, ws_mean, ws_var);
    bn_silu_kernel<<<(NEL + 255) / 256, 256, 0, stream>>>(ws_y, out, ws_mean, ws_var, gamma, beta);
}

extern "C" void kernel_launch(void* const* d_in, const int* in_sizes, int n_in,
                              void* d_out, int out_size, void* d_ws, size_t ws_size,
                              hipStream_t stream) {
    (void)in_sizes; (void)n_in; (void)out_size; (void)ws_size;
    const float* RGB = (const float*)d_in[0];
    const float* Tin = (const float*)d_in[1];
    const float *dw_[4], *db_[4], *ow_[4], *ob_[4], *gm_[4], *bt_[4];
    int i = 2;
    for (int d = 0; d < 4; ++d) {
        dw_[d] = (const float*)d_in[i++]; db_[d] = (const float*)d_in[i++];
        ow_[d] = (const float*)d_in[i++]; ob_[d] = (const float*)d_in[i++];
        gm_[d] = (const float*)d_in[i++]; bt_[d] = (const float*)d_in[i++];
    }
    const float* mlp_dw = (const float*)d_in[i++];
    const float* mlp_db = (const float*)d_in[i++];
    const float* mlp_uw = (const float*)d_in[i++];
    const float* mlp_ub = (const float*)d_in[i++];
    const float* dwc_w  = (const float*)d_in[i++];
    const float* dwc_b  = (const float*)d_in[i++];
    const float* ec_dw  = (const float*)d_in[i++];
    const float* ec_db  = (const float*)d_in[i++];
    const float* ec_uw  = (const float*)d_in[i++];
    const float* ec_ub  = (const float*)d_in[i++];
    const float* srw    = (const float*)d_in[i++];
    const float* srb    = (const float*)d_in[i++];
    const float* stw    = (const float*)d_in[i++];
    const float* stb    = (const float*)d_in[i++];

    float* ws = (float*)d_ws;
    size_t off = 0;
    float* buf_v    = ws + off; off += (size_t)BATCH * CK * HW;     // 10.6M
    float* buf_om   = ws + off; off += (size_t)BATCH * 27 * HW;
    float* buf_y    = ws + off; off += (size_t)NEL;
    float* buf_mean = ws + off; off += CCH;
    float* buf_var  = ws + off; off += CCH;
    float* buf_R    = ws + off; off += (size_t)NEL;
    float* buf_T    = ws + off; off += (size_t)NEL;
    float* buf_poolR= ws + off; off += BATCH * CCH;
    float* buf_poolT= ws + off; off += BATCH * CCH;
    float* buf_h    = ws + off; off += BATCH * 2 * CCH;
    float* buf_ry   = ws + off; off += BATCH * CCH;
    float* buf_ty   = ws + off; off += BATCH * CCH;
    float* buf_cross= ws + off; off += BATCH * CCH;
    float* buf_dw   = ws + off; off += (size_t)BATCH * 2 * CCH * HW;
    float* buf_mx   = ws + off; off += BATCH * 2 * CCH;
    float* buf_h2   = ws + off; off += BATCH * 32;
    float* buf_fg   = ws + off; off += BATCH * 2 * CCH;
    float* buf_nR   = ws + off; off += (size_t)NEL;
    float* buf_nT   = ws + off; off += (size_t)NEL;
    float* buf_att  = ws + off; off += BATCH * 2 * HW;

    // DCN1 / DCN2
    run_dcn(RGB, dw_[0], db_[0], ow_[0], ob_[0], gm_[0], bt_[0],
            buf_R, buf_v, buf_om, buf_y, buf_mean, buf_var, stream);
    run_dcn(Tin, dw_[1], db_[1], ow_[1], ob_[1], gm_[1], bt_[1],
            buf_T, buf_v, buf_om, buf_y, buf_mean, buf_var, stream);

    // shared MLP pool -> ry / ty
    mean_pool_kernel<<<(BATCH * CCH + 255) / 256, 256, 0, stream>>>(buf_R, buf_poolR);
    linear_kernel<<<(BATCH * 512 + 255) / 256, 256, 0, stream>>>(buf_poolR, mlp_dw, mlp_db, buf_h, BATCH, 256, 512, 1);
    linear_kernel<<<(BATCH * 256 + 255) / 256, 256, 0, stream>>>(buf_h, mlp_uw, mlp_ub, buf_ry, BATCH, 512, 256, 0);
    mean_pool_kernel<<<(BATCH * CCH + 255) / 256, 256, 0, stream>>>(buf_T, buf_poolT);
    linear_kernel<<<(BATCH * 512 + 255) / 256, 256, 0, stream>>>(buf_poolT, mlp_dw, mlp_db, buf_h, BATCH, 256, 512, 1);
    linear_kernel<<<(BATCH * 256 + 255) / 256, 256, 0, stream>>>(buf_h, mlp_uw, mlp_ub, buf_ty, BATCH, 512, 256, 0);

    norm_cross_kernel<<<BATCH, 256, 0, stream>>>(buf_ry, buf_ty, buf_cross);

    // 7x7 grouped conv on implicit concat [Re,Te], then channel SE
    dwconv_kernel<<<(BATCH * 2 * CCH * HW + 255) / 256, 256, 0, stream>>>(buf_R, buf_T, buf_cross, dwc_w, dwc_b, buf_dw);
    max_pool_kernel<<<(BATCH * 2 * CCH + 255) / 256, 256, 0, stream>>>(buf_dw, buf_mx);
    linear_kernel<<<(BATCH * 32 + 255) / 256, 256, 0, stream>>>(buf_mx, ec_dw, ec_db, buf_h2, BATCH, 512, 32, 1);
    linear_kernel<<<(BATCH * 512 + 255) / 256, 256, 0, stream>>>(buf_h2, ec_uw, ec_ub, buf_fg, BATCH, 32, 512, 2);

    fuse_nrt_kernel<<<(NEL + 255) / 256, 256, 0, stream>>>(buf_R, buf_T, buf_cross, buf_fg, buf_nR, buf_nT);
    sse_att_kernel<<<(BATCH * HW + 255) / 256, 256, 0, stream>>>(buf_nR, buf_nT, srw, srb, stw, stb, buf_att);
    apply_att_kernel<<<(NEL + 255) / 256, 256, 0, stream>>>(buf_nR, buf_nT, buf_att);

    // DCN3 / DCN4 straight into output regions, then sum
    float* outR = (float*)d_out;
    float* outT = outR + (size_t)NEL;
    float* outS = outT + (size_t)NEL;
    run_dcn(buf_nR, dw_[2], db_[2], ow_[2], ob_[2], gm_[2], bt_[2],
            outR, buf_v, buf_om, buf_y, buf_mean, buf_var, stream);
    run_dcn(buf_nT, dw_[3], db_[3], ow_[3], ob_[3], gm_[3], bt_[3],
            outT, buf_v, buf_om, buf_y, buf_mean, buf_var, stream);
    add_kernel<<<(NEL + 255) / 256, 256, 0, stream>>>(outR, outT, outS, NEL);
}